// ProposalFilter_2757369004237
// MI455X (gfx1250) — compile-verified
//
#include <hip/hip_runtime.h>
#include <stdint.h>

#define N        6000
#define NPAD     8192          // next pow2 >= N for bitonic sort
#define T        1024          // 32 wave32 waves per workgroup
#define WORDS    192           // ceil(6144/32) keep-mask words (N padded to 6144 bits)
#define CHUNKS   188           // ceil(N/32)
#define SOAPAD   6016          // N padded to 32-row chunk boundary
#define IOU_THR  0.7f

typedef float    v2f  __attribute__((ext_vector_type(2)));
typedef float    v8f  __attribute__((ext_vector_type(8)));
typedef unsigned u32x4 __attribute__((ext_vector_type(4)));
typedef int      i32x8 __attribute__((ext_vector_type(8)));
typedef int      i32x4 __attribute__((ext_vector_type(4)));

extern "C" __global__ void __launch_bounds__(T)
proposal_filter_nms_kernel(const float* __restrict__ proposals,
                           const float* __restrict__ scores,
                           float* __restrict__ outBoxes,
                           float* __restrict__ outKeep)
{
    extern __shared__ char smemRaw[];
    float*    raw  = (float*)smemRaw;            // 24000 f32 : raw boxes (TDM target)
    float*    keys = raw + N * 4;                // 8192  f32 : sort keys
    unsigned* idxs = (unsigned*)(keys + NPAD);   // 8192  u32 : sort payload
    float*    x1S  = (float*)(idxs + NPAD);      // 6016 each : sorted SoA
    float*    y1S  = x1S + SOAPAD;
    float*    x2S  = y1S + SOAPAD;
    float*    y2S  = x2S + SOAPAD;
    float*    arS  = y2S + SOAPAD;
    unsigned* keep    = (unsigned*)(arS + SOAPAD); // 192 u32 : keep bitmask
    unsigned* rowmask = keep + WORDS;              // 32*192 u32 : per-chunk suppression rows

    const int b    = blockIdx.x;
    const int tid  = threadIdx.x;
    const int lane = tid & 31;
    const int wave = tid >> 5;

    // ---------------- Phase 0: TDM async tensor load of raw boxes into LDS ----------
    if (tid < 32) {
        uint64_t ga   = (uint64_t)(uintptr_t)(proposals + (size_t)b * N * 4);
        uint32_t ldsA = (uint32_t)(uintptr_t)raw;   // low 32 bits of generic LDS addr = LDS offset
        const uint32_t elems = (uint32_t)(N * 4);   // 24000 f32, one contiguous row
        u32x4 g0;
        g0[0] = 1u;                                            // count=1, user D#
        g0[1] = ldsA;                                          // lds_addr
        g0[2] = (uint32_t)(ga & 0xFFFFFFFFu);                  // global_addr[31:0]
        g0[3] = (uint32_t)((ga >> 32) & 0x01FFFFFFu) | (2u << 30); // global_addr[56:32] | type=2
        i32x8 g1;
        g1[0] = (int)(2u << 16);                               // workgroup_mask=0, data_size=4B
        g1[1] = (int)((elems & 0xFFFFu) << 16);                // tensor_dim0[15:0]
        g1[2] = (int)((elems >> 16) | (1u << 16));             // tensor_dim0[31:16] | tensor_dim1=1
        g1[3] = (int)((elems & 0xFFFFu) << 16);                // tile_dim0 = 24000
        g1[4] = 1;                                             // tile_dim1=1, tile_dim2=0
        g1[5] = (int)elems;                                    // tensor_dim0_stride[31:0]
        g1[6] = 0;
        g1[7] = 0;
        i32x4 g2 = {0, 0, 0, 0};
        i32x4 g3 = {0, 0, 0, 0};
        i32x8 g4 = {0, 0, 0, 0, 0, 0, 0, 0};                   // extra group (unused, 2D tensor)
        __builtin_amdgcn_tensor_load_to_lds(g0, g1, g2, g3, g4, 0);
        __builtin_amdgcn_s_wait_tensorcnt(0);
    }

    // ---------------- Phase 1: load score keys (descending sort, stable tiebreak) ----
    for (int i = tid; i < NPAD; i += T) {
        keys[i] = (i < N) ? scores[(size_t)b * N * 2 + 2 * i + 1] : -__builtin_huge_valf();
        idxs[i] = (unsigned)i;
    }
    __syncthreads();

    // ---------------- Phase 2: bitonic sort in LDS (91 steps) ------------------------
    for (unsigned k = 2; k <= NPAD; k <<= 1) {
        for (unsigned j = k >> 1; j > 0; j >>= 1) {
            for (unsigned i = tid; i < NPAD; i += T) {
                unsigned ixj = i ^ j;
                if (ixj > i) {
                    float    ka = keys[i], kb = keys[ixj];
                    unsigned ia = idxs[i], ib = idxs[ixj];
                    // 'before' = a precedes b in final order (score desc, idx asc)
                    bool before = (ka > kb) || (ka == kb && ia < ib);
                    bool descB  = ((i & k) == 0);
                    if (descB ? !before : before) {
                        keys[i] = kb; keys[ixj] = ka;
                        idxs[i] = ib; idxs[ixj] = ia;
                    }
                }
            }
            __syncthreads();
        }
    }

    // ---------------- Phase 3: gather sorted SoA + areas, init keep bits -------------
    for (int i = tid; i < SOAPAD; i += T) {
        float bx1 = 0.f, by1 = 0.f, bx2 = 0.f, by2 = 0.f, a = 0.f;
        if (i < N) {
            unsigned s = idxs[i];
            bx1 = raw[s * 4 + 0]; by1 = raw[s * 4 + 1];
            bx2 = raw[s * 4 + 2]; by2 = raw[s * 4 + 3];
            a = (bx2 - bx1) * (by2 - by1);
        }
        x1S[i] = bx1; y1S[i] = by1; x2S[i] = bx2; y2S[i] = by2; arS[i] = a;
    }
    for (int w = tid; w < WORDS; w += T)
        keep[w] = (w < 187) ? 0xFFFFFFFFu : ((w == 187) ? 0x0000FFFFu : 0u); // N=6000 bits
    __syncthreads();

    // ---------------- Phase 4: chunked NMS ------------------------------------------
    for (int c = 0; c < CHUNKS; ++c) {
        const int i0 = c * 32;
        // 4a: clear this chunk's 32 suppression rows
        for (int w = tid; w < 32 * WORDS; w += T) rowmask[w] = 0u;
        __syncthreads();

        // 4b: build suppression bits with WMMA 16x16 tiles.
        // union base:  D[m][n] = area_row[m]*1 + 1*area_col[n]  via v_wmma_f32_16x16x4_f32
        const int t0    = 2 * c;              // first 16-wide col tile (cols >= i0 only)
        const int units = 2 * (375 - t0);     // 2 row-halves x col tiles
        for (int u = wave; u < units; u += 32) {
            const int rh    = u & 1;
            const int j0    = (t0 + (u >> 1)) * 16;   // col base (multiple of 16)
            const int rbase = i0 + rh * 16;           // row base
            v2f A, B;
            A.x = (lane < 16) ? arS[rbase + lane] : 0.f;  // A: K=0 col = area_row
            A.y = (lane < 16) ? 1.0f : 0.f;               // A: K=1 col = ones
            B.x = (lane < 16) ? 1.0f : 0.f;               // B: K=0 row = ones
            B.y = (lane < 16) ? arS[j0 + lane] : 0.f;     // B: K=1 row = area_col
            v8f Cz = {};
            v8f D = __builtin_amdgcn_wmma_f32_16x16x4_f32(
                false, A, false, B, (short)0, Cz, false, false);

            const int nIdx = j0 + (lane & 15);
            const float cx1 = x1S[nIdx], cy1 = y1S[nIdx];
            const float cx2 = x2S[nIdx], cy2 = y2S[nIdx];

            unsigned myBits = 0u; int myRow = -1;
            #pragma unroll
            for (int v = 0; v < 8; ++v) {
                const int row = rbase + v + ((lane & 16) ? 8 : 0);
                const float rx1 = x1S[row], ry1 = y1S[row];
                const float rx2 = x2S[row], ry2 = y2S[row];
                float iw = fminf(rx2, cx2) - fmaxf(rx1, cx1);
                float ih = fminf(ry2, cy2) - fmaxf(ry1, cy1);
                iw = fmaxf(iw, 0.f); ih = fmaxf(ih, 0.f);
                const float inter = iw * ih;
                const float uni   = D[v] - inter;            // area_i + area_j - inter
                const bool pred   = (inter >= IOU_THR * uni) && (nIdx > row);
                const unsigned bal = __builtin_amdgcn_ballot_w32(pred);
                // ballot bits 0..15 = row v (lanes 0-15), bits 16..31 = row v+8
                if (lane == v)     { myBits = bal & 0xFFFFu; myRow = rh * 16 + v; }
                if (lane == v + 8) { myBits = bal >> 16;     myRow = rh * 16 + v + 8; }
            }
            if (myRow >= 0)
                atomicOr(&rowmask[myRow * WORDS + (j0 >> 5)], myBits << (j0 & 16));
        }
        __syncthreads();

        // 4c: wave-synchronous sequential resolve (single wave, no block barriers)
        if (tid < 32) {
            volatile unsigned* keepV = keep;
            volatile unsigned* rmV   = rowmask;
            for (int r = 0; r < 32; ++r) {
                const int i = i0 + r;
                if (i >= N) break;
                const unsigned kw = keepV[i >> 5];            // uniform broadcast read
                if ((kw >> (i & 31)) & 1u) {
                    for (int w = c + lane; w < WORDS; w += 32)
                        keepV[w] &= ~rmV[r * WORDS + w];
                }
            }
        }
        __syncthreads();
    }

    // ---------------- Phase 5: write outputs ----------------------------------------
    for (int i = tid; i < N; i += T) {
        const unsigned kw = keep[i >> 5];
        const float kf = ((kw >> (i & 31)) & 1u) ? 1.0f : 0.0f;
        const size_t ob = ((size_t)b * N + i) * 4;
        outBoxes[ob + 0] = x1S[i] * kf;
        outBoxes[ob + 1] = y1S[i] * kf;
        outBoxes[ob + 2] = x2S[i] * kf;
        outBoxes[ob + 3] = y2S[i] * kf;
        outKeep[(size_t)b * N + i] = kf;
    }
}

extern "C" void kernel_launch(void* const* d_in, const int* in_sizes, int n_in,
                              void* d_out, int out_size, void* d_ws, size_t ws_size,
                              hipStream_t stream) {
    (void)n_in; (void)out_size; (void)d_ws; (void)ws_size;
    const float* proposals = (const float*)d_in[0];
    const float* scores    = (const float*)d_in[1];
    float* out = (float*)d_out;
    const int B = in_sizes[0] / (N * 4);           // 4
    float* keepOut = out + (size_t)B * N * 4;      // keep mask follows boxes, flat
    const size_t shmem =
        (size_t)(N * 4 + NPAD + NPAD + 5 * SOAPAD + WORDS + 32 * WORDS) * sizeof(float);
    proposal_filter_nms_kernel<<<B, T, shmem, stream>>>(proposals, scores, out, keepOut);
}